// ArgMaxTop_22497038696878
// MI455X (gfx1250) — compile-verified
//
#include <hip/hip_runtime.h>
#include <hip/hip_bf16.h>

#define B_ 8
#define S_ 2048
#define C_ 32000
#define K_ 8
#define L_ (B_ * S_ * K_)   // 131072 : flattened top-k stream length

typedef __attribute__((address_space(3))) float4 lds_float4;

// s_wait_asynccnt with a literal immediate (token-pasted so it is always an
// assembler constant).
#define WAIT_ASYNC(n) asm volatile("s_wait_asynccnt " #n ::: "memory")

// ---------------------------------------------------------------------------
// Kernel 0: init workspace (sums = 0, first_pos = L). Runs every call.
// ---------------------------------------------------------------------------
__global__ __launch_bounds__(256) void argmaxtop_init(float* __restrict__ sums,
                                                      int* __restrict__ first_pos) {
  int i = blockIdx.x * 256 + threadIdx.x;
  if (i < B_ * C_) sums[i] = 0.0f;
  if (i < C_)      first_pos[i] = L_;
}

// Sorted-descending top-8 insert kept fully in VGPRs (unrolled -> cndmask chain).
__device__ __forceinline__ void insert8(float (&v)[8], int (&g)[8], float x, int idx) {
  if (x <= v[7]) return;
  v[7] = x; g[7] = idx;
#pragma unroll
  for (int j = 7; j >= 1; --j) {
    if (v[j] > v[j - 1]) {
      float tv = v[j]; v[j] = v[j - 1]; v[j - 1] = tv;
      int   tg = g[j]; g[j] = g[j - 1]; g[j - 1] = tg;
    }
  }
}

// ---------------------------------------------------------------------------
// Kernel 1: one wave32 per row. Stream the row through LDS with the CDNA5
// async global->LDS path (4-deep pipeline), keep per-lane top-8 in registers,
// wave-merge with wave32 butterflies, scatter with atomics.
// ---------------------------------------------------------------------------
__global__ __launch_bounds__(256) void argmaxtop_topk(const float* __restrict__ in,
                                                      float* __restrict__ sums,
                                                      int* __restrict__ first_pos) {
  const int lane = threadIdx.x & 31;
  const int wave = threadIdx.x >> 5;
  const int row  = blockIdx.x * 8 + wave;          // row in [0, B*S)
  const int b    = row >> 11;                      // row / S_  (S_ = 2048)
  const float* rowPtr = in + (size_t)row * C_;

  // 8 waves * 4 buffers * 32 lanes * 16B = 16 KB staging (of 320 KB/WGP)
  __shared__ float4 stage[8][4][32];

  const int NIT = C_ / 128;                        // 250 chunks of 32 float4s

  float v[8]; int g[8];
#pragma unroll
  for (int j = 0; j < 8; ++j) { v[j] = -__builtin_inff(); g[j] = 0x7fffffff; }

  auto issue = [&](int i) {
    unsigned voff = (unsigned)i * 512u + (unsigned)lane * 16u;
    asm volatile("global_load_async_to_lds_b128 %0, %1, %2"
                 :
                 : "v"((lds_float4*)&stage[wave][i & 3][lane]),
                   "v"(voff),
                   "s"(rowPtr)
                 : "memory");
  };
  auto consume = [&](int i) {
    float4 x = stage[wave][i & 3][lane];           // ds_load_b128
    // one guard for all four elements: rarely true once v[7] has risen
    float m4 = fmaxf(fmaxf(x.x, x.y), fmaxf(x.z, x.w));
    if (m4 > v[7]) {
      int baseIdx = i * 128 + lane * 4;            // class index of x.x
      insert8(v, g, x.x, baseIdx + 0);
      insert8(v, g, x.y, baseIdx + 1);
      insert8(v, g, x.z, baseIdx + 2);
      insert8(v, g, x.w, baseIdx + 3);
    }
  };

  // prologue: fill 3 of 4 pipeline slots
  issue(0); issue(1); issue(2);

  // main: keep 4 chunks in flight; wait until <=3 outstanding => chunk i done
  const int mainEnd = NIT - 3;                     // 247
  for (int i = 0; i < mainEnd; ++i) {
    issue(i + 3);
    WAIT_ASYNC(3);
    consume(i);
  }
  // drain: waits 2,1,0 guarantee chunks NIT-3, NIT-2, NIT-1 respectively
  WAIT_ASYNC(2); consume(NIT - 3);
  WAIT_ASYNC(1); consume(NIT - 2);
  WAIT_ASYNC(0); consume(NIT - 1);

  // Wave32 merge: 8 rounds of butterfly arg-max over (value desc, class asc)
  // -- matches jax.lax.top_k ordering (ties -> smaller index first).
  // Winner lane shifts its queue down (static register indices only).
  const float NEG_INF = -__builtin_inff();
  for (int r = 0; r < 8; ++r) {
    float mv = v[0]; int mg = g[0];
    float bv = mv;  int bg = mg;
#pragma unroll
    for (int off = 16; off > 0; off >>= 1) {
      float ov = __shfl_xor(bv, off, 32);
      int   og = __shfl_xor(bg, off, 32);
      if (ov > bv || (ov == bv && og < bg)) { bv = ov; bg = og; }
    }
    if (mv == bv && mg == bg) {                    // unique winner (g unique per row)
#pragma unroll
      for (int j = 0; j < 7; ++j) { v[j] = v[j + 1]; g[j] = g[j + 1]; }
      v[7] = NEG_INF; g[7] = 0x7fffffff;
    }
    if (lane == 0) {
      atomicAdd(&sums[(size_t)b * C_ + bg], bv);
      atomicMin(&first_pos[bg], row * K_ + r);     // flattened stream position
    }
  }
}

// ---------------------------------------------------------------------------
// Kernel 2: per-batch argmax of mean with reference tie-break:
// maximize masked mean; among exact ties minimize first-occurrence position.
// ---------------------------------------------------------------------------
__global__ __launch_bounds__(256) void argmaxtop_final(const float* __restrict__ sums,
                                                       const int* __restrict__ first_pos,
                                                       int* __restrict__ out) {
  const int b   = blockIdx.x;
  const int tid = threadIdx.x;
  float bv = -__builtin_inff(); int bf = 0x7fffffff; int bc = 0;
  for (int c = tid; c < C_; c += 256) {
    int f = first_pos[c];
    if (f >= L_) continue;                         // class never appears -> -inf
    float m = sums[(size_t)b * C_ + c] * (1.0f / (float)S_);  // exact: * 2^-11
    if (m > bv || (m == bv && f < bf)) { bv = m; bf = f; bc = c; }
  }
  __shared__ float sv[256]; __shared__ int sf[256]; __shared__ int sc[256];
  sv[tid] = bv; sf[tid] = bf; sc[tid] = bc;
  __syncthreads();
  for (int s2 = 128; s2 > 0; s2 >>= 1) {
    if (tid < s2) {
      if (sv[tid + s2] > sv[tid] ||
          (sv[tid + s2] == sv[tid] && sf[tid + s2] < sf[tid])) {
        sv[tid] = sv[tid + s2]; sf[tid] = sf[tid + s2]; sc[tid] = sc[tid + s2];
      }
    }
    __syncthreads();
  }
  if (tid == 0) out[b] = sc[0];
}

// ---------------------------------------------------------------------------
extern "C" void kernel_launch(void* const* d_in, const int* in_sizes, int n_in,
                              void* d_out, int out_size, void* d_ws, size_t ws_size,
                              hipStream_t stream) {
  (void)in_sizes; (void)n_in; (void)out_size; (void)ws_size;
  const float* in  = (const float*)d_in[0];
  int*   out       = (int*)d_out;                        // reference output is int32 [B]
  float* sums      = (float*)d_ws;                       // B*C floats = 1,024,000 B
  int*   first_pos = (int*)((char*)d_ws + (size_t)B_ * C_ * sizeof(float)); // C ints

  argmaxtop_init <<<(B_ * C_ + 255) / 256, 256, 0, stream>>>(sums, first_pos);
  argmaxtop_topk <<<(B_ * S_) / 8,        256, 0, stream>>>(in, sums, first_pos);
  argmaxtop_final<<<B_,                   256, 0, stream>>>(sums, first_pos, out);
}